// Layer_6219112644903
// MI455X (gfx1250) — compile-verified
//
#include <hip/hip_runtime.h>

// LIF spiking-neuron forward scan:
//   x: [B, T, C, H, W] f32, thresh: [C] f32 (broadcast over 1,C,1,1)
//   for t: mem = 0.5*mem + x[:,t]; spike = mem > thresh; mem = spike ? 0 : mem
//   out: spikes [B, T, C, H, W] f32
//
// Pure HBM stream (256 MB total, > 192 MB L2) -> bandwidth-bound, ~11 us floor
// at 23.3 TB/s. One 256-thread block (8 wave32 waves) handles one (b,c) plane
// of H*W = 1024 floats, 4 floats/thread via b128 vector ops with non-temporal
// hints. Shapes are fixed by the harness, so the t-plane stride is a
// compile-time constant (1 MB): t*stride*4B fits the signed 24-bit IOFFSET,
// letting all 8 loads (and 8 stores) share a single base-address VGPR pair
// with immediate offsets -- no per-plane 64-bit address adds.

typedef float v4f __attribute__((ext_vector_type(4)));

#define TAU 0.5f

constexpr int TSTEPS = 8;
constexpr int CCH    = 256;          // channels
constexpr int HW     = 32 * 32;      // pixels per plane
constexpr int PLANE  = CCH * HW;     // elements between consecutive t planes

__global__ __launch_bounds__(256) void lif_scan_kernel(
    const float* __restrict__ x,
    const float* __restrict__ thresh,
    float* __restrict__ out)
{
    const int c = blockIdx.x;            // channel  (uniform, from TTMP)
    const int b = blockIdx.y;            // batch    (uniform, from TTMP)

    // thresh[c] is uniform across the block -> scalar load + broadcast
    const float th = thresh[c];

    const int base = (b * TSTEPS * CCH + c) * HW + (int)threadIdx.x * 4;
    const float* xb = x   + base;        // one VGPR-pair address for all loads
    float*       ob = out + base;        // one VGPR-pair address for all stores

    // Issue all 8 independent t-plane loads up front: 8 b128 loads in flight
    // per thread, non-temporal (single-pass stream, don't pollute L2). Each
    // uses the same address register + immediate offset t*1MB.
    v4f v[TSTEPS];
#pragma unroll
    for (int t = 0; t < TSTEPS; ++t) {
        v[t] = __builtin_nontemporal_load((const v4f*)(xb + t * PLANE));
    }

    // 8-step recurrence, 4 independent lanes of ILP per thread.
    // Per element per step: v_fma (tau=0.5 inline const), v_cmp_gt,
    // v_cndmask(spike), v_cndmask(reset) -- both selects reuse the same VCC.
    v4f mem;
    mem.x = 0.0f; mem.y = 0.0f; mem.z = 0.0f; mem.w = 0.0f;

#pragma unroll
    for (int t = 0; t < TSTEPS; ++t) {
        mem = mem * TAU + v[t];

        v4f spike;
        spike.x = (mem.x > th) ? 1.0f : 0.0f;
        spike.y = (mem.y > th) ? 1.0f : 0.0f;
        spike.z = (mem.z > th) ? 1.0f : 0.0f;
        spike.w = (mem.w > th) ? 1.0f : 0.0f;

        // hard reset: mem = spike ? 0 : mem  (== mem * (1 - spike))
        mem.x = (mem.x > th) ? 0.0f : mem.x;
        mem.y = (mem.y > th) ? 0.0f : mem.y;
        mem.z = (mem.z > th) ? 0.0f : mem.z;
        mem.w = (mem.w > th) ? 0.0f : mem.w;

        __builtin_nontemporal_store(spike, (v4f*)(ob + t * PLANE));
    }
}

extern "C" void kernel_launch(void* const* d_in, const int* in_sizes, int n_in,
                              void* d_out, int out_size, void* d_ws, size_t ws_size,
                              hipStream_t stream) {
    const float* x      = (const float*)d_in[0];
    const float* thresh = (const float*)d_in[1];
    float*       out    = (float*)d_out;

    const int B = in_sizes[0] / (TSTEPS * CCH * HW);   // 16

    // One block per (b,c) image plane; 256 threads * 4 floats = H*W = 1024.
    dim3 grid(CCH, B);
    dim3 block(HW / 4);
    lif_scan_kernel<<<grid, block, 0, stream>>>(x, thresh, out);
}